// DetectorKmeans_61649960567014
// MI455X (gfx1250) — compile-verified
//
#include <hip/hip_runtime.h>

typedef float v2f __attribute__((ext_vector_type(2)));
typedef float v8f __attribute__((ext_vector_type(8)));

#define N_PTS 500000
#define DIM   64
#define KC    128

// LDS B layout for V_WMMA_F32_16X16X4_F32, indexed so the 8 fragments used by
// one kk-step are adjacent (256 B apart -> ds_load_2addr_b64 pairs):
//   bLds[(kk*8 + j)*32 + lane] = { centers[col][k], centers[col][k+1] }
//   col = j*16 + (lane&15),  k = 4*kk + 2*(lane>>4)
// (B KxN = 4x16: lanes 0-15 hold K=0,1 ; lanes 16-31 hold K=2,3; N = lane%16)

__global__ __launch_bounds__(256)
void gmm_density_kernel(const float* __restrict__ X,
                        const float* __restrict__ centers,
                        const float* __restrict__ vars_,
                        const float* __restrict__ prs,
                        const float* __restrict__ threshold,
                        float* __restrict__ out)
{
    __shared__ v2f   bLds[16 * 8 * 32];   // 32 KB: centers pre-swizzled to WMMA-B layout
    __shared__ float alphaLds[KC];
    __shared__ float gammaLds[KC];
    __shared__ float betaLds[KC];

    const int tid   = threadIdx.x;
    const int lane  = tid & 31;
    const int wave  = tid >> 5;
    const int laneM = lane & 15;
    const int hi    = lane >> 4;

    const int rowBase = blockIdx.x * 128 + wave * 16;

    // ---- issue streaming X loads FIRST so they overlap the LDS staging ----
    int rowA = rowBase + laneM;
    if (rowA > N_PTS - 1) rowA = N_PTS - 1;          // clamp: EXEC must stay all-ones for WMMA
    const float* xrow = X + (size_t)rowA * DIM + 2 * hi;

    v2f a[16];
    #pragma unroll
    for (int kk = 0; kk < 16; ++kk)
        a[kk] = *(const v2f*)(xrow + 4 * kk);

    // ---- stage centers into WMMA-B layout (once per block, L2-resident source) ----
    #pragma unroll
    for (int i = 0; i < 16; ++i) {
        int e   = tid + i * 256;           // 4096 float2 entries
        int ln  = e & 31;
        int j   = (e >> 5) & 7;
        int kk  = e >> 8;
        int col = j * 16 + (ln & 15);
        int k   = 4 * kk + 2 * (ln >> 4);
        v2f b;
        b.x = centers[col * DIM + k];
        b.y = centers[col * DIM + k + 1];
        bLds[e] = b;
    }
    // ---- per-center affine constants: alpha = -0.5/var, gamma = 1/var,
    //      beta = alpha*||c||^2 + log(prs) ----
    if (tid < KC) {
        float c2 = 0.f;
        #pragma unroll
        for (int k = 0; k < DIM; ++k) {
            float c = centers[tid * DIM + k];
            c2 = fmaf(c, c, c2);
        }
        float v     = vars_[tid];
        float alpha = -0.5f / v;
        alphaLds[tid] = alpha;
        gammaLds[tid] = 1.0f / v;
        betaLds[tid]  = alpha * c2 + __logf(prs[tid]);
    }

    // x2 per row while waiting: lane l and l^16 each hold half of row laneM
    float x2p = 0.f;
    #pragma unroll
    for (int kk = 0; kk < 16; ++kk) {
        x2p = fmaf(a[kk].x, a[kk].x, x2p);
        x2p = fmaf(a[kk].y, a[kk].y, x2p);
    }
    float x2full = x2p + __shfl_xor(x2p, 16, 32);
    // D-matrix rows live at VGPR index v: row = v + 8*hi. Fetch x2 for those rows.
    float x2v[8];
    #pragma unroll
    for (int v = 0; v < 8; ++v)
        x2v[v] = __shfl(x2full, v + 8 * hi, 32);

    __syncthreads();

    // ---- 8 independent WMMA accumulation chains (j inner => no RAW back-to-back) ----
    v8f acc[8];
    #pragma unroll
    for (int j = 0; j < 8; ++j)
        acc[j] = (v8f){0.f, 0.f, 0.f, 0.f, 0.f, 0.f, 0.f, 0.f};

    const v2f* bw = bLds + lane;
    #pragma unroll
    for (int kk = 0; kk < 16; ++kk) {
        #pragma unroll
        for (int j = 0; j < 8; ++j) {
            acc[j] = __builtin_amdgcn_wmma_f32_16x16x4_f32(
                false, a[kk], false, bw[(kk * 8 + j) * 32], (short)0, acc[j], false, false);
        }
    }

    // ---- pass 1: affine map cross -> logits in place; per-lane row max ----
    float m[8];
    #pragma unroll
    for (int v = 0; v < 8; ++v) m[v] = -3.0e38f;

    #pragma unroll
    for (int j = 0; j < 8; ++j) {
        int   col   = j * 16 + laneM;
        float alpha = alphaLds[col];
        float gamma = gammaLds[col];
        float beta  = betaLds[col];
        #pragma unroll
        for (int v = 0; v < 8; ++v) {
            float t = fmaf(gamma, acc[j][v], fmaf(alpha, x2v[v], beta));
            acc[j][v] = t;
            m[v] = fmaxf(m[v], t);
        }
    }

    // ---- pass 2: max-butterfly across the 16 lanes sharing a row half ----
    #pragma unroll
    for (int w = 1; w < 16; w <<= 1) {
        #pragma unroll
        for (int v = 0; v < 8; ++v)
            m[v] = fmaxf(m[v], __shfl_xor(m[v], w, 32));
    }

    // ---- pass 3: one exp per logit against the global row max ----
    float s[8];
    #pragma unroll
    for (int v = 0; v < 8; ++v) s[v] = 0.f;
    #pragma unroll
    for (int j = 0; j < 8; ++j) {
        #pragma unroll
        for (int v = 0; v < 8; ++v)
            s[v] += __expf(acc[j][v] - m[v]);
    }

    // ---- pass 4: sum-butterfly (no rescaling needed: m is already global) ----
    #pragma unroll
    for (int w = 1; w < 16; w <<= 1) {
        #pragma unroll
        for (int v = 0; v < 8; ++v)
            s[v] += __shfl_xor(s[v], w, 32);
    }

    // ---- store: lanes with laneM<8 write row = rowBase + laneM + 8*hi ----
    float thr  = threshold[0];
    float dens = 0.f;
    #pragma unroll
    for (int v = 0; v < 8; ++v)
        if (laneM == v) dens = m[v] + __logf(s[v]);

    int rowOut = rowBase + laneM + 8 * hi;
    if (laneM < 8 && rowOut < N_PTS)
        out[rowOut] = dens - thr;
}

extern "C" void kernel_launch(void* const* d_in, const int* in_sizes, int n_in,
                              void* d_out, int out_size, void* d_ws, size_t ws_size,
                              hipStream_t stream) {
    const float* X       = (const float*)d_in[0];
    const float* centers = (const float*)d_in[1];
    const float* vars_   = (const float*)d_in[2];
    const float* prs     = (const float*)d_in[3];
    const float* thr     = (const float*)d_in[4];
    float* out           = (float*)d_out;
    (void)in_sizes; (void)n_in; (void)out_size; (void)d_ws; (void)ws_size;

    int nblocks = (N_PTS + 127) / 128;   // 128 rows per block (8 waves x 16 rows)
    gmm_density_kernel<<<nblocks, 256, 0, stream>>>(X, centers, vars_, prs, thr, out);
}